// Model_53137335386831
// MI455X (gfx1250) — compile-verified
//
#include <hip/hip_runtime.h>
#include <math.h>

// ---------------- problem constants ----------------
#define Bn   2
#define F0c  540
#define F1c  960
#define D0c  270
#define D1c  480
#define P0c  135
#define P1c  240
#define MU   1.0e-4f
#define INV_MU 1.0e4f
#define TAUc 2.0e-4f
#define NITER 5

#define NPLANE (F0c*F1c)           // 518400
#define NPIX   (Bn*F0c*F1c)        // 1036800 (one channel plane set)
#define NRr    (3*Bn*F0c*F1c)      // 3110400
// state layout is channel-major CBIJ: t = ((c*Bn + b)*F0 + i)*F1 + j

typedef __attribute__((ext_vector_type(2))) float v2f;
typedef __attribute__((ext_vector_type(8))) float v8f;

#define WMMA4(ACC, Aop, Bop) \
    ACC = __builtin_amdgcn_wmma_f32_16x16x4_f32(false, Aop, false, Bop, (short)0, ACC, false, false)

static __device__ __forceinline__ v2f ld2(const float* p) { return *(const v2f*)p; }

// =====================================================================
// Matmul-DFT passes, split-complex, transform axis contiguous (stride 1).
//   out[row][k] = sum_n in[row][n] * W[k*N + n]        (W symmetric)
// One wave -> 32x32 output tile (2x2 blocking of 16x16 WMMA tiles).
// Output indexing: r -> (r/rowsBo, r%rowsBo):
//   off = (r/rowsBo)*sAo + (r%rowsBo)*sBo + col*esO    (fold transposes here)
// =====================================================================

// ---- complex -> complex (4 WMMA combos, 16 wmma / K-step) ----
__global__ __launch_bounds__(32)
void dft_cc(const float* __restrict__ inre, const float* __restrict__ inim,
            float* __restrict__ outre, float* __restrict__ outim,
            const float* __restrict__ Wre, const float* __restrict__ Wim,
            int N, int rows, int inRS,
            int rowsBo, long long sAo, long long sBo, long long esO)
{
    const int lane = threadIdx.x;
    const int m    = lane & 15;
    const int kh2  = (lane >> 4) * 2;

    int r0 = blockIdx.x * 32 + m;      if (r0 >= rows) r0 = rows - 1;
    int r1 = blockIdx.x * 32 + 16 + m; if (r1 >= rows) r1 = rows - 1;
    int c0 = blockIdx.y * 32 + m;      if (c0 >= N)    c0 = N - 1;
    int c1 = blockIdx.y * 32 + 16 + m; if (c1 >= N)    c1 = N - 1;

    const float* a0r = inre + (long long)r0 * inRS + kh2;
    const float* a0i = inim + (long long)r0 * inRS + kh2;
    const float* a1r = inre + (long long)r1 * inRS + kh2;
    const float* a1i = inim + (long long)r1 * inRS + kh2;
    const float* b0r = Wre + (long long)c0 * N + kh2;
    const float* b0i = Wim + (long long)c0 * N + kh2;
    const float* b1r = Wre + (long long)c1 * N + kh2;
    const float* b1i = Wim + (long long)c1 * N + kh2;

    v8f Crr[2][2] = {}, Cii[2][2] = {}, Cri[2][2] = {}, Cir[2][2] = {};

#pragma unroll 4
    for (int kk = 0; kk < N; kk += 4) {
        v2f Ar[2], Ai[2], Br[2], Bi[2];
        Ar[0] = ld2(a0r + kk); Ai[0] = ld2(a0i + kk);
        Ar[1] = ld2(a1r + kk); Ai[1] = ld2(a1i + kk);
        Br[0] = ld2(b0r + kk); Bi[0] = ld2(b0i + kk);
        Br[1] = ld2(b1r + kk); Bi[1] = ld2(b1i + kk);
#pragma unroll
        for (int x = 0; x < 2; ++x)
#pragma unroll
            for (int q = 0; q < 2; ++q) {
                WMMA4(Crr[x][q], Ar[x], Br[q]);
                WMMA4(Cii[x][q], Ai[x], Bi[q]);
                WMMA4(Cri[x][q], Ar[x], Bi[q]);
                WMMA4(Cir[x][q], Ai[x], Br[q]);
            }
    }

    const int mb = (lane < 16) ? 0 : 8;
    const int cols[2] = {c0, c1};
#pragma unroll
    for (int x = 0; x < 2; ++x)
#pragma unroll
        for (int q = 0; q < 2; ++q)
            for (int g = 0; g < 8; ++g) {
                int M = blockIdx.x * 32 + x * 16 + mb + g;
                if (M >= rows) M = rows - 1;   // duplicate write, identical value
                long long o = (long long)(M / rowsBo) * sAo +
                              (long long)(M % rowsBo) * sBo +
                              (long long)cols[q] * esO;
                outre[o] = Crr[x][q][g] - Cii[x][q][g];
                outim[o] = Cri[x][q][g] + Cir[x][q][g];
            }
}

// ---- real -> complex (first forward pass; 8 wmma / K-step) ----
__global__ __launch_bounds__(32)
void dft_rc(const float* __restrict__ in,
            float* __restrict__ outre, float* __restrict__ outim,
            const float* __restrict__ Wre, const float* __restrict__ Wim,
            int N, int rows, int inRS,
            int rowsBo, long long sAo, long long sBo, long long esO)
{
    const int lane = threadIdx.x;
    const int m    = lane & 15;
    const int kh2  = (lane >> 4) * 2;

    int r0 = blockIdx.x * 32 + m;      if (r0 >= rows) r0 = rows - 1;
    int r1 = blockIdx.x * 32 + 16 + m; if (r1 >= rows) r1 = rows - 1;
    int c0 = blockIdx.y * 32 + m;      if (c0 >= N)    c0 = N - 1;
    int c1 = blockIdx.y * 32 + 16 + m; if (c1 >= N)    c1 = N - 1;

    const float* a0 = in + (long long)r0 * inRS + kh2;
    const float* a1 = in + (long long)r1 * inRS + kh2;
    const float* b0r = Wre + (long long)c0 * N + kh2;
    const float* b0i = Wim + (long long)c0 * N + kh2;
    const float* b1r = Wre + (long long)c1 * N + kh2;
    const float* b1i = Wim + (long long)c1 * N + kh2;

    v8f Cre[2][2] = {}, Cim[2][2] = {};

#pragma unroll 4
    for (int kk = 0; kk < N; kk += 4) {
        v2f Ar[2], Br[2], Bi[2];
        Ar[0] = ld2(a0 + kk); Ar[1] = ld2(a1 + kk);
        Br[0] = ld2(b0r + kk); Bi[0] = ld2(b0i + kk);
        Br[1] = ld2(b1r + kk); Bi[1] = ld2(b1i + kk);
#pragma unroll
        for (int x = 0; x < 2; ++x)
#pragma unroll
            for (int q = 0; q < 2; ++q) {
                WMMA4(Cre[x][q], Ar[x], Br[q]);
                WMMA4(Cim[x][q], Ar[x], Bi[q]);
            }
    }

    const int mb = (lane < 16) ? 0 : 8;
    const int cols[2] = {c0, c1};
#pragma unroll
    for (int x = 0; x < 2; ++x)
#pragma unroll
        for (int q = 0; q < 2; ++q)
            for (int g = 0; g < 8; ++g) {
                int M = blockIdx.x * 32 + x * 16 + mb + g;
                if (M >= rows) M = rows - 1;
                long long o = (long long)(M / rowsBo) * sAo +
                              (long long)(M % rowsBo) * sBo +
                              (long long)cols[q] * esO;
                outre[o] = Cre[x][q][g];
                outim[o] = Cim[x][q][g];
            }
}

// ---- complex -> real (last inverse pass, real part only; 8 wmma / K-step) ----
__global__ __launch_bounds__(32)
void dft_cr(const float* __restrict__ inre, const float* __restrict__ inim,
            float* __restrict__ out,
            const float* __restrict__ Wre, const float* __restrict__ Wim,
            int N, int rows, int inRS,
            int rowsBo, long long sAo, long long sBo, long long esO)
{
    const int lane = threadIdx.x;
    const int m    = lane & 15;
    const int kh2  = (lane >> 4) * 2;

    int r0 = blockIdx.x * 32 + m;      if (r0 >= rows) r0 = rows - 1;
    int r1 = blockIdx.x * 32 + 16 + m; if (r1 >= rows) r1 = rows - 1;
    int c0 = blockIdx.y * 32 + m;      if (c0 >= N)    c0 = N - 1;
    int c1 = blockIdx.y * 32 + 16 + m; if (c1 >= N)    c1 = N - 1;

    const float* a0r = inre + (long long)r0 * inRS + kh2;
    const float* a0i = inim + (long long)r0 * inRS + kh2;
    const float* a1r = inre + (long long)r1 * inRS + kh2;
    const float* a1i = inim + (long long)r1 * inRS + kh2;
    const float* b0r = Wre + (long long)c0 * N + kh2;
    const float* b0i = Wim + (long long)c0 * N + kh2;
    const float* b1r = Wre + (long long)c1 * N + kh2;
    const float* b1i = Wim + (long long)c1 * N + kh2;

    v8f Crr[2][2] = {}, Cii[2][2] = {};

#pragma unroll 4
    for (int kk = 0; kk < N; kk += 4) {
        v2f Ar[2], Ai[2], Br[2], Bi[2];
        Ar[0] = ld2(a0r + kk); Ai[0] = ld2(a0i + kk);
        Ar[1] = ld2(a1r + kk); Ai[1] = ld2(a1i + kk);
        Br[0] = ld2(b0r + kk); Bi[0] = ld2(b0i + kk);
        Br[1] = ld2(b1r + kk); Bi[1] = ld2(b1i + kk);
#pragma unroll
        for (int x = 0; x < 2; ++x)
#pragma unroll
            for (int q = 0; q < 2; ++q) {
                WMMA4(Crr[x][q], Ar[x], Br[q]);
                WMMA4(Cii[x][q], Ai[x], Bi[q]);
            }
    }

    const int mb = (lane < 16) ? 0 : 8;
    const int cols[2] = {c0, c1};
#pragma unroll
    for (int x = 0; x < 2; ++x)
#pragma unroll
        for (int q = 0; q < 2; ++q)
            for (int g = 0; g < 8; ++g) {
                int M = blockIdx.x * 32 + x * 16 + mb + g;
                if (M >= rows) M = rows - 1;
                long long o = (long long)(M / rowsBo) * sAo +
                              (long long)(M % rowsBo) * sBo +
                              (long long)cols[q] * esO;
                out[o] = Crr[x][q][g] - Cii[x][q][g];
            }
}

// ---------------- DFT matrix fill (split re/im; W symmetric) ----------------
__global__ void fillW(float* __restrict__ Wre, float* __restrict__ Wim,
                      int N, double sgn, float scale)
{
    int t = blockIdx.x * blockDim.x + threadIdx.x;
    if (t >= N * N) return;
    int n = t / N, k = t % N;
    double ang = 6.283185307179586 * (double)(((long long)n * k) % N) / (double)N;
    Wre[t] = (float)cos(ang) * scale;
    Wim[t] = (float)(sgn * sin(ang)) * scale;
}

// ---------------- H precompute ----------------
__global__ void pack_h(float* __restrict__ o, const float* __restrict__ h)
{
    int t = blockIdx.x * blockDim.x + threadIdx.x;
    if (t >= NPLANE) return;
    int i = t / F1c, j = t % F1c;
    float v = 0.f;
    if (i >= P0c && i < P0c + D0c && j >= P1c && j < P1c + D1c)
        v = h[(i - P0c) * D1c + (j - P1c)];
    o[t] = v;
}

// raw is fft2(hpad) in split (j,i) layout: t = j*540 + i
__global__ void finalizeH(float2* __restrict__ H, float* __restrict__ Sm,
                          const float* __restrict__ rawre, const float* __restrict__ rawim)
{
    int t = blockIdx.x * blockDim.x + threadIdx.x;
    if (t >= NPLANE) return;
    int i = t % F0c, j = t / F0c;
    float rr = rawre[t], ri = rawim[t];
    float s = ((i + j) & 1) ? -1.f : 1.f;        // fold ifftshift of hpad
    H[i * F1c + j] = make_float2(s * rr, s * ri);
    float hth = rr * rr + ri * ri;
    float ltl = 4.f - 2.f * cosf(6.28318530718f * (float)i / (float)F0c)
                    - 2.f * cosf(6.28318530718f * (float)j / (float)F1c);
    Sm[i * F1c + j] = 1.f / (MU * hth + MU * ltl + MU);
}

// ---------------- pointwise spectral multipliers (split, CBJI layout) -------
// buffers between the two 540-passes are laid out (c,b,j,i): t -> i = t%540,
// j = (t/540)%960
__global__ void cmulH(float* __restrict__ bre, float* __restrict__ bim,
                      const float2* __restrict__ H, int conj, int n)
{
    int t = blockIdx.x * blockDim.x + threadIdx.x;
    if (t >= n) return;
    int i = t % F0c, j = (t / F0c) % F1c;
    float2 hv = H[i * F1c + j];
    if (conj) hv.y = -hv.y;
    float xr = bre[t], xi = bim[t];
    bre[t] = xr * hv.x - xi * hv.y;
    bim[t] = xr * hv.y + xi * hv.x;
}

__global__ void cmulS(float* __restrict__ bre, float* __restrict__ bim,
                      const float* __restrict__ S, int n)
{
    int t = blockIdx.x * blockDim.x + threadIdx.x;
    if (t >= n) return;
    int i = t % F0c, j = (t / F0c) % F1c;
    float s = S[i * F1c + j];
    bre[t] *= s;
    bim[t] *= s;
}

// ---------------- ADMM pointwise kernels (state layout CBIJ) ----------------
__global__ void k_u(const float* __restrict__ sk, const float* __restrict__ a21,
                    const float* __restrict__ a22, float* __restrict__ u1,
                    float* __restrict__ u2, int n)
{
    int t = blockIdx.x * blockDim.x + threadIdx.x;
    if (t >= n) return;
    int j = t % F1c, i = (t / F1c) % F0c;
    float v = 0.f, hh = 0.f;
    if (i < F0c - 1) v  = -(sk[t + F1c] - sk[t]) + a21[t] * INV_MU;
    if (j < F1c - 1) hh = -(sk[t + 1]   - sk[t]) + a22[t] * INV_MU;
    float sq = v * v + hh * hh;
    float mag = (sq > 0.f) ? sqrtf(sq) : 0.f;
    float magt = fmaxf(mag - TAUc, 0.f);
    float mm = magt / (magt + TAUc);
    if (i < F0c - 1) u1[t] = v * mm;
    if (j < F1c - 1) u2[t] = hh * mm;
}

__global__ void k_vkp(float* __restrict__ vkp, const float* __restrict__ a1,
                      const float* __restrict__ Hs, const float* __restrict__ y, int n)
{
    int t = blockIdx.x * blockDim.x + threadIdx.x;
    if (t >= n) return;
    int j = t % F1c, i = (t / F1c) % F0c, b = (t / NPLANE) % Bn, c = t / NPIX;
    bool inside = (i >= P0c && i < P0c + D0c && j >= P1c && j < P1c + D1c);
    float cty = inside ? y[((b * D0c + (i - P0c)) * D1c + (j - P1c)) * 3 + c] : 0.f;
    float vm = inside ? (1.f / (1.f + MU)) : (1.f / MU);
    vkp[t] = vm * (a1[t] + MU * Hs[t] + cty);
}

__global__ void k_wkp(float* __restrict__ wkp, const float* __restrict__ a3,
                      const float* __restrict__ sk, int n)
{
    int t = blockIdx.x * blockDim.x + threadIdx.x;
    if (t >= n) return;
    wkp[t] = fmaxf(a3[t] * INV_MU + sk[t], 0.f);
}

__global__ void k_pre(float* __restrict__ o, const float* __restrict__ vkp,
                      const float* __restrict__ a1, int n)
{
    int t = blockIdx.x * blockDim.x + threadIdx.x;
    if (t >= n) return;
    o[t] = vkp[t] - a1[t] * INV_MU;
}

__global__ void k_num(float* __restrict__ num, const float* __restrict__ wkp,
                      const float* __restrict__ a3, const float* __restrict__ hadj,
                      const float* __restrict__ u1, const float* __restrict__ a21,
                      const float* __restrict__ u2, const float* __restrict__ a22, int n)
{
    int t = blockIdx.x * blockDim.x + threadIdx.x;
    if (t >= n) return;
    int j = t % F1c, i = (t / F1c) % F0c;
    float ta, tb;
    if (i == 0)           ta =  (u1[t] - a21[t] * INV_MU);
    else if (i < F0c - 1) ta =  (u1[t] - a21[t] * INV_MU) - (u1[t - F1c] - a21[t - F1c] * INV_MU);
    else                  ta = -(u1[t - F1c] - a21[t - F1c] * INV_MU);
    if (j == 0)           tb =  (u2[t] - a22[t] * INV_MU);
    else if (j < F1c - 1) tb =  (u2[t] - a22[t] * INV_MU) - (u2[t - 1] - a22[t - 1] * INV_MU);
    else                  tb = -(u2[t - 1] - a22[t - 1] * INV_MU);
    num[t] = MU * wkp[t] - a3[t] + MU * hadj[t] + MU * (ta + tb);
}

__global__ void k_a13(float* __restrict__ a1, float* __restrict__ a3,
                      const float* __restrict__ Hs, const float* __restrict__ vkp,
                      const float* __restrict__ sk, const float* __restrict__ wkp, int n)
{
    int t = blockIdx.x * blockDim.x + threadIdx.x;
    if (t >= n) return;
    a1[t] += MU * (Hs[t] - vkp[t]);
    a3[t] += MU * (sk[t] - wkp[t]);
}

__global__ void k_a2(float* __restrict__ a21, float* __restrict__ a22,
                     const float* __restrict__ sk, const float* __restrict__ u1,
                     const float* __restrict__ u2, int n)
{
    int t = blockIdx.x * blockDim.x + threadIdx.x;
    if (t >= n) return;
    int j = t % F1c, i = (t / F1c) % F0c;
    if (i < F0c - 1) a21[t] += MU * (-(sk[t + F1c] - sk[t]) - u1[t]);
    if (j < F1c - 1) a22[t] += MU * (-(sk[t + 1]   - sk[t]) - u2[t]);
}

// output is (b,i,j,c); state is CBIJ
__global__ void k_crop(float* __restrict__ out, const float* __restrict__ sk, int n)
{
    int t = blockIdx.x * blockDim.x + threadIdx.x;
    if (t >= n) return;
    int c = t % 3, j = (t / 3) % D1c, i = (t / (3 * D1c)) % D0c, b = t / (3 * D1c * D0c);
    out[t] = sk[(long long)c * NPIX + (long long)b * NPLANE +
                (long long)(i + P0c) * F1c + (j + P1c)];
}

// =====================================================================
// Host orchestration
// =====================================================================
extern "C" void kernel_launch(void* const* d_in, const int* in_sizes, int n_in,
                              void* d_out, int out_size, void* d_ws, size_t ws_size,
                              hipStream_t stream)
{
    (void)in_sizes; (void)n_in; (void)out_size; (void)ws_size;
    const float* y = (const float*)d_in[0];
    const float* h = (const float*)d_in[1];
    float* outp = (float*)d_out;

    char* base = (char*)d_ws;
    size_t off = 0;
    auto grab = [&](size_t bytes) -> void* {
        void* p = base + off;
        off += (bytes + 255) & ~(size_t)255;
        return p;
    };

    float* cAre = (float*)grab((size_t)NRr * 4);
    float* cAim = (float*)grab((size_t)NRr * 4);
    float* cBre = (float*)grab((size_t)NRr * 4);
    float* cBim = (float*)grab((size_t)NRr * 4);
    float* W960fre = (float*)grab((size_t)960 * 960 * 4);
    float* W960fim = (float*)grab((size_t)960 * 960 * 4);
    float* W960ire = (float*)grab((size_t)960 * 960 * 4);
    float* W960iim = (float*)grab((size_t)960 * 960 * 4);
    float* W540fre = (float*)grab((size_t)540 * 540 * 4);
    float* W540fim = (float*)grab((size_t)540 * 540 * 4);
    float* W540ire = (float*)grab((size_t)540 * 540 * 4);
    float* W540iim = (float*)grab((size_t)540 * 540 * 4);
    float2* Hf = (float2*)grab((size_t)NPLANE * 8);
    float*  Sm = (float*) grab((size_t)NPLANE * 4);
    float* skA = (float*)grab((size_t)NRr * 4);
    float* skB = (float*)grab((size_t)NRr * 4);
    float* HsA = (float*)grab((size_t)NRr * 4);
    float* HsB = (float*)grab((size_t)NRr * 4);
    float* a1  = (float*)grab((size_t)NRr * 4);
    float* a3  = (float*)grab((size_t)NRr * 4);
    float* vkp = (float*)grab((size_t)NRr * 4);
    float* wkp = (float*)grab((size_t)NRr * 4);
    float* num = (float*)grab((size_t)NRr * 4);
    float* hbuf= (float*)grab((size_t)NRr * 4);
    float* u1  = (float*)grab((size_t)NRr * 4);
    float* u2  = (float*)grab((size_t)NRr * 4);
    float* a21 = (float*)grab((size_t)NRr * 4);
    float* a22 = (float*)grab((size_t)NRr * 4);

    const int TB = 256;
    dim3 pw((NRr + TB - 1) / TB), blk(TB);
    dim3 pwPlane((NPLANE + TB - 1) / TB);

    hipMemsetAsync(skA, 0, (size_t)NRr * 4, stream);
    hipMemsetAsync(HsA, 0, (size_t)NRr * 4, stream);
    hipMemsetAsync(a1,  0, (size_t)NRr * 4, stream);
    hipMemsetAsync(a3,  0, (size_t)NRr * 4, stream);
    hipMemsetAsync(a21, 0, (size_t)NRr * 4, stream);
    hipMemsetAsync(a22, 0, (size_t)NRr * 4, stream);
    hipMemsetAsync(u1,  0, (size_t)NRr * 4, stream);
    hipMemsetAsync(u2,  0, (size_t)NRr * 4, stream);

    // DFT matrices (fwd: e^{-i}, inv: e^{+i}/N); channel-axis FFT cancels exactly
    fillW<<<dim3((960*960 + TB - 1) / TB), blk, 0, stream>>>(W960fre, W960fim, 960, -1.0, 1.0f);
    fillW<<<dim3((960*960 + TB - 1) / TB), blk, 0, stream>>>(W960ire, W960iim, 960,  1.0, 1.0f / 960.f);
    fillW<<<dim3((540*540 + TB - 1) / TB), blk, 0, stream>>>(W540fre, W540fim, 540, -1.0, 1.0f);
    fillW<<<dim3((540*540 + TB - 1) / TB), blk, 0, stream>>>(W540ire, W540iim, 540,  1.0, 1.0f / 540.f);

    // ---- H = fft2(hpad) * (-1)^{i+j};  Smult from |H|^2 + closed-form LtL ----
    float* hpR = cBre;  // transient
    pack_h<<<pwPlane, blk, 0, stream>>>(hpR, h);
    // j-pass on real hpad (i,j) -> split (j,i)
    dft_rc<<<dim3(17, 30), 32, 0, stream>>>(hpR, cAre, cAim, W960fre, W960fim,
                                            960, 540, 960, 1, 1LL, 0LL, 540LL);
    // i-pass (j,i) -> (j,i)
    dft_cc<<<dim3(30, 17), 32, 0, stream>>>(cAre, cAim, cBre, cBim, W540fre, W540fim,
                                            540, 960, 540, 1, 540LL, 0LL, 1LL);
    finalizeH<<<pwPlane, blk, 0, stream>>>(Hf, Sm, cBre, cBim);

    // ---- fft2 -> multiplier -> ifft2, per channel (shifts cancel; c-FFT cancels)
    auto chain = [&](const float* src, float* dst, int mode /*0:H 1:conjH 2:Smult*/) {
        // fwd j-pass: real CBIJ -> split CBJI
        dft_rc<<<dim3(102, 30), 32, 0, stream>>>(src, cAre, cAim, W960fre, W960fim,
                                                 960, 3240, 960, 540, 518400LL, 1LL, 540LL);
        // fwd i-pass: CBJI -> CBJI (contiguous stores)
        dft_cc<<<dim3(180, 17), 32, 0, stream>>>(cAre, cAim, cBre, cBim, W540fre, W540fim,
                                                 540, 5760, 540, 1, 540LL, 0LL, 1LL);
        if (mode == 2) cmulS<<<pw, blk, 0, stream>>>(cBre, cBim, Sm, NRr);
        else           cmulH<<<pw, blk, 0, stream>>>(cBre, cBim, Hf, mode, NRr);
        // inv i-pass: CBJI -> CBIJ
        dft_cc<<<dim3(180, 17), 32, 0, stream>>>(cBre, cBim, cAre, cAim, W540ire, W540iim,
                                                 540, 5760, 540, 960, 518400LL, 1LL, 960LL);
        // inv j-pass: CBIJ -> real CBIJ (real part only)
        dft_cr<<<dim3(102, 30), 32, 0, stream>>>(cAre, cAim, dst, W960ire, W960iim,
                                                 960, 3240, 960, 1, 960LL, 0LL, 1LL);
    };

    float *sk = skA, *skN = skB, *Hs = HsA, *HsN = HsB;
    for (int it = 0; it < NITER; ++it) {
        k_u  <<<pw, blk, 0, stream>>>(sk, a21, a22, u1, u2, NRr);
        k_vkp<<<pw, blk, 0, stream>>>(vkp, a1, Hs, y, NRr);
        k_wkp<<<pw, blk, 0, stream>>>(wkp, a3, sk, NRr);
        k_pre<<<pw, blk, 0, stream>>>(hbuf, vkp, a1, NRr);
        chain(hbuf, hbuf, 1);                              // hadj(vkp - a1/mu1)
        k_num<<<pw, blk, 0, stream>>>(num, wkp, a3, hbuf, u1, a21, u2, a22, NRr);
        chain(num, skN, 2);                                // skp
        chain(skN, HsN, 0);                                // Hskp = hfor(skp)
        k_a13<<<pw, blk, 0, stream>>>(a1, a3, HsN, vkp, skN, wkp, NRr);
        k_a2 <<<pw, blk, 0, stream>>>(a21, a22, skN, u1, u2, NRr);
        float* t;
        t = sk; sk = skN; skN = t;
        t = Hs; Hs = HsN; HsN = t;
    }
    k_crop<<<dim3((Bn*D0c*D1c*3 + TB - 1) / TB), blk, 0, stream>>>(outp, sk, Bn*D0c*D1c*3);
}